// PointGNN_11931419149118
// MI455X (gfx1250) — compile-verified
//
#include <hip/hip_runtime.h>
#include <hip/hip_bf16.h>

#if defined(__HIP_DEVICE_COMPILE__)
#if !__has_builtin(__builtin_amdgcn_wmma_f32_16x16x4_f32)
#error "gfx1250 f32 WMMA builtin __builtin_amdgcn_wmma_f32_16x16x4_f32 not available (device pass)"
#endif
#endif

typedef __attribute__((ext_vector_type(2))) float v2f;
typedef __attribute__((ext_vector_type(8))) float v8f;

#define NB   4
#define MP   384
#define PTOT (NB * MP)      // 1536 points
#define TT   3
#define RTH  0.05f
#define WPR  12             // mask words per point row (384/32)

// ---------------------------------------------------------------------------
// One 16x16 fp32 WMMA output tile, accumulating over K (multiple of 4).
//   A: 16 x K activations (LDS, row stride lda)
//   B: K x 16 weights    (global, row stride ldb, pre-offset to column tile)
// Layout per ISA 7.12.2 (32-bit A 16x4 / B 4x16 / C 16x16, wave32).
// ---------------------------------------------------------------------------
__device__ __forceinline__ v8f wmma_acc_f32(const float* __restrict__ A, int lda,
                                            const float* __restrict__ B, int ldb,
                                            int K, v8f acc, int lane) {
  const int row  = lane & 15;          // M (A) / N (B) index for this lane
  const int koff = (lane >> 4) << 1;   // lanes 0-15: K=0,1  lanes 16-31: K=2,3
  A += row * lda + koff;
  B += koff * ldb + row;
#if defined(__HIP_DEVICE_COMPILE__)
  for (int k = 0; k < K; k += 4) {
    v2f a, b;
    a.x = A[k];
    a.y = A[k + 1];
    b.x = B[k * ldb];
    b.y = B[(k + 1) * ldb];
    acc = __builtin_amdgcn_wmma_f32_16x16x4_f32(false, a, false, b, (short)0,
                                                acc, false, false);
  }
#endif
  return acc;
}

// ---------------------------------------------------------------------------
// state <- x
// ---------------------------------------------------------------------------
__global__ void k_init_state(const float* __restrict__ x, float* __restrict__ s, int n) {
  int i = blockIdx.x * blockDim.x + threadIdx.x;
  if (i < n) s[i] = x[i];
}

// ---------------------------------------------------------------------------
// Adjacency bitmasks: mask[(n*MP+i)*WPR + w] bit b  <=>  ||x_i - x_j||^2 < RTH,
// j = 32*w + b.  Computed once (adjacency depends on x, not state).
// ---------------------------------------------------------------------------
__global__ void k_adj(const float* __restrict__ x, unsigned* __restrict__ mask) {
  int tid = blockIdx.x * blockDim.x + threadIdx.x;
  if (tid >= PTOT * WPR) return;
  int w  = tid % WPR;
  int ni = tid / WPR;
  int n  = ni / MP;
  int i  = ni % MP;
  const float xi0 = x[(n * MP + i) * 3 + 0];
  const float xi1 = x[(n * MP + i) * 3 + 1];
  const float xi2 = x[(n * MP + i) * 3 + 2];
  unsigned bits = 0u;
  const int jb = w * 32;
  for (int b = 0; b < 32; ++b) {
    int j = jb + b;
    float d0 = x[(n * MP + j) * 3 + 0] - xi0;
    float d1 = x[(n * MP + j) * 3 + 1] - xi1;
    float d2 = x[(n * MP + j) * 3 + 2] - xi2;
    if (d0 * d0 + d1 * d1 + d2 * d2 < RTH) bits |= (1u << b);
  }
  mask[tid] = bits;
}

// ---------------------------------------------------------------------------
// Edge MLP, collapsed to per-source-node:  F[p] = MLP_f([0,0,0, state[p]]).
// Block = 16 points, 256 threads (8 waves); wave w owns output columns
// [16w,16w+16).  Layer1 (3->64) on VALU, layers 2 (64->128) and 3 (128->128)
// on fp32 WMMA.
// ---------------------------------------------------------------------------
__global__ void __launch_bounds__(256) k_mlp_f(
    const float* __restrict__ state,
    const float* __restrict__ W1, const float* __restrict__ b1,   // (6,64), (64,)
    const float* __restrict__ W2, const float* __restrict__ b2,   // (64,128), (128,)
    const float* __restrict__ W3, const float* __restrict__ b3,   // (128,128), (128,)
    float* __restrict__ F) {
  __shared__ float s[16][3];
  __shared__ float h1[16][64 + 4];     // padded stride: conflict-free A gathers
  __shared__ float h2[16][128 + 4];

  const int tid = threadIdx.x;
  const int p0  = blockIdx.x * 16;

  if (tid < 48) {
    int r = tid / 3, d = tid % 3;
    s[r][d] = state[(p0 + r) * 3 + d];
  }
  __syncthreads();

  // layer 1: rows 0..2 of W1 multiply the all-zero 'rel' slot -> only rows 3..5
  for (int idx = tid; idx < 16 * 64; idx += 256) {
    int r = idx >> 6, c = idx & 63;
    float v = b1[c] + s[r][0] * W1[3 * 64 + c]
                    + s[r][1] * W1[4 * 64 + c]
                    + s[r][2] * W1[5 * 64 + c];
    h1[r][c] = fmaxf(v, 0.0f);
  }
  __syncthreads();

  const int wave = tid >> 5;
  const int lane = tid & 31;
  const int col  = lane & 15;
  const int rb   = (lane >> 4) << 3;

  // layer 2: 16x128 = 8 waves x (16x16 tile), K = 64
  {
    const int c0 = wave * 16;
    const float bv = b2[c0 + col];
    v8f acc = {bv, bv, bv, bv, bv, bv, bv, bv};
    acc = wmma_acc_f32(&h1[0][0], 64 + 4, W2 + c0, 128, 64, acc, lane);
    for (int g = 0; g < 8; ++g) h2[rb + g][c0 + col] = fmaxf(acc[g], 0.0f);
  }
  __syncthreads();

  // layer 3: 16x128, K = 128 -> F (final relu)
  {
    const int c0 = wave * 16;
    const float bv = b3[c0 + col];
    v8f acc = {bv, bv, bv, bv, bv, bv, bv, bv};
    acc = wmma_acc_f32(&h2[0][0], 128 + 4, W3 + c0, 128, 128, acc, lane);
    for (int g = 0; g < 8; ++g)
      F[(p0 + rb + g) * 128 + c0 + col] = fmaxf(acc[g], 0.0f);
  }
}

// ---------------------------------------------------------------------------
// agg[n,i,c] = max over adjacent j of F[n,j,c]; init 0 matches the dense
// reference max (masked entries are exactly 0 and F >= 0 post-relu).
// One block per (n,i); thread = channel c (coalesced F reads).
// ---------------------------------------------------------------------------
__global__ void __launch_bounds__(128) k_agg(const float* __restrict__ F,
                                             const unsigned* __restrict__ mask,
                                             float* __restrict__ agg) {
  const int ni = blockIdx.x;
  const int c  = threadIdx.x;
  const int n  = ni / MP;
  float acc = 0.0f;
  for (int w = 0; w < WPR; ++w) {
    unsigned bits = mask[ni * WPR + w];   // wave-uniform scalar load
    while (bits) {
      int b = __builtin_ctz(bits);
      bits &= bits - 1;
      int j = w * 32 + b;
      acc = fmaxf(acc, F[(n * MP + j) * 128 + c]);
    }
  }
  agg[ni * 128 + c] = acc;
}

// ---------------------------------------------------------------------------
// Node MLP + residual: out = state_in + relu(MLP_g(agg)).
// Block = 16 points. Layer1 (128->64): waves 0..3 WMMA; layer2 (64->32):
// waves 0..1 WMMA; layer3 (32->3) + residual on VALU.
// ---------------------------------------------------------------------------
__global__ void __launch_bounds__(256) k_mlp_g(
    const float* __restrict__ agg,
    const float* __restrict__ si, float* __restrict__ so,
    const float* __restrict__ W1, const float* __restrict__ b1,   // (128,64)
    const float* __restrict__ W2, const float* __restrict__ b2,   // (64,32)
    const float* __restrict__ W3, const float* __restrict__ b3) { // (32,3)
  __shared__ float a[16][128 + 4];
  __shared__ float h1[16][64 + 4];
  __shared__ float h2[16][32 + 4];

  const int tid = threadIdx.x;
  const int p0  = blockIdx.x * 16;

  for (int idx = tid; idx < 16 * 128; idx += 256) {
    int r = idx >> 7, c = idx & 127;
    a[r][c] = agg[(p0 + r) * 128 + c];
  }
  __syncthreads();

  const int wave = tid >> 5;
  const int lane = tid & 31;
  const int col  = lane & 15;
  const int rb   = (lane >> 4) << 3;

  if (wave < 4) {                       // wave-uniform: EXEC all-ones in wave
    const int c0 = wave * 16;
    const float bv = b1[c0 + col];
    v8f acc = {bv, bv, bv, bv, bv, bv, bv, bv};
    acc = wmma_acc_f32(&a[0][0], 128 + 4, W1 + c0, 64, 128, acc, lane);
    for (int g = 0; g < 8; ++g) h1[rb + g][c0 + col] = fmaxf(acc[g], 0.0f);
  }
  __syncthreads();

  if (wave < 2) {
    const int c0 = wave * 16;
    const float bv = b2[c0 + col];
    v8f acc = {bv, bv, bv, bv, bv, bv, bv, bv};
    acc = wmma_acc_f32(&h1[0][0], 64 + 4, W2 + c0, 32, 64, acc, lane);
    for (int g = 0; g < 8; ++g) h2[rb + g][c0 + col] = fmaxf(acc[g], 0.0f);
  }
  __syncthreads();

  if (tid < 48) {
    int r = tid / 3, d = tid % 3;
    float v = b3[d];
    for (int k = 0; k < 32; ++k) v += h2[r][k] * W3[k * 3 + d];
    so[(p0 + r) * 3 + d] = si[(p0 + r) * 3 + d] + fmaxf(v, 0.0f);
  }
}

// ---------------------------------------------------------------------------
extern "C" void kernel_launch(void* const* d_in, const int* in_sizes, int n_in,
                              void* d_out, int out_size, void* d_ws, size_t ws_size,
                              hipStream_t stream) {
  const float* x   = (const float*)d_in[0];
  // d_in[1..6] = hW1..hb3 : provably unused by the reference (delta is dead).
  const float* fW1 = (const float*)d_in[7];
  const float* fb1 = (const float*)d_in[8];
  const float* fW2 = (const float*)d_in[9];
  const float* fb2 = (const float*)d_in[10];
  const float* fW3 = (const float*)d_in[11];
  const float* fb3 = (const float*)d_in[12];
  const float* gW1 = (const float*)d_in[13];
  const float* gb1 = (const float*)d_in[14];
  const float* gW2 = (const float*)d_in[15];
  const float* gb2 = (const float*)d_in[16];
  const float* gW3 = (const float*)d_in[17];
  const float* gb3 = (const float*)d_in[18];

  float* ws    = (float*)d_ws;
  float* state = ws;                         // PTOT*3
  float* F     = state + PTOT * 3;           // PTOT*128
  float* agg   = F + PTOT * 128;             // PTOT*128
  unsigned* mask = (unsigned*)(agg + PTOT * 128);  // PTOT*WPR u32

  k_init_state<<<(PTOT * 3 + 255) / 256, 256, 0, stream>>>(x, state, PTOT * 3);
  k_adj<<<(PTOT * WPR + 255) / 256, 256, 0, stream>>>(x, mask);

  for (int t = 0; t < TT; ++t) {
    k_mlp_f<<<PTOT / 16, 256, 0, stream>>>(
        state,
        fW1 + t * 6 * 64,   fb1 + t * 64,
        fW2 + t * 64 * 128, fb2 + t * 128,
        fW3 + t * 128 * 128, fb3 + t * 128,
        F);
    k_agg<<<PTOT, 128, 0, stream>>>(F, mask, agg);
    const float* sin = state;
    float* sout = (t == TT - 1) ? (float*)d_out : state;
    k_mlp_g<<<PTOT / 16, 256, 0, stream>>>(
        agg, sin, sout,
        gW1 + t * 128 * 64, gb1 + t * 64,
        gW2 + t * 64 * 32,  gb2 + t * 32,
        gW3 + t * 32 * 3,   gb3 + t * 3);
  }
}